// DifferentiableRenderer_10471130268177
// MI455X (gfx1250) — compile-verified
//
#include <hip/hip_runtime.h>
#include <hip/hip_bf16.h>

typedef float v2f __attribute__((ext_vector_type(2)));
typedef float v8f __attribute__((ext_vector_type(8)));

#define GRID_N   40
#define GRID_SZ  (GRID_N * GRID_N * GRID_N)   // 64000 cells
#define NVOX     (32 * 32 * 32)               // 32768 source voxels
#define BLOCK    256                          // 8 waves (wave32)

// One workgroup per batch element.
//   Phase 0: zero the 40^3 u16 occupancy grid in LDS (125 KB, fits CDNA5 WGP LDS).
//   Phase 1: coords = scaled @ R via V_WMMA_F32_16X16X4_F32.
//            A = R^T padded to 16x4 (loop invariant), B = 16 voxel coord columns.
//            Lane n<16 gets (cx,cy,cz) of voxel n in d[0..2] -> clamp/trunc -> LDS scatter.
//   Phase 2: per-(x,y) ray march along z with exclusive transmittance.
__global__ __launch_bounds__(BLOCK)
void DifferentiableRenderer_10471130268177_kernel(
    const float* __restrict__ camR,   // [B,3,3]
    const float* __restrict__ absb,   // [32,32,32,1] logits
    const float* __restrict__ attn,   // [32,32,32,1] logits
    float* __restrict__ out)          // [B,40,40,1]
{
    extern __shared__ unsigned short st[];    // GRID_SZ entries: 0 = empty, else src_index+1

    const int b    = blockIdx.x;
    const int tid  = threadIdx.x;
    const int lane = tid & 31;
    const int wave = tid >> 5;

    // ---- Phase 0: clear occupancy grid --------------------------------------
    for (int i = tid; i < GRID_SZ; i += BLOCK) st[i] = 0;
    __syncthreads();

    const float* R = camR + b * 9;            // R[k][m] at R[k*3 + m]

    // ---- Build loop-invariant A operand: A[m][k] = R[k][m] (zero padded) ----
    // 32-bit 16x4 A layout: lanes 0-15 -> row m=lane, VGPR0=K0, VGPR1=K1;
    //                       lanes 16-31 -> row m=lane-16, VGPR0=K2, VGPR1=K3.
    const int m = lane & 15;
    v2f a;
    if (lane < 16) {
        a.x = (m < 3) ? R[0 * 3 + m] : 0.0f;  // A[m][0] = R[0][m]
        a.y = (m < 3) ? R[1 * 3 + m] : 0.0f;  // A[m][1] = R[1][m]
    } else {
        a.x = (m < 3) ? R[2 * 3 + m] : 0.0f;  // A[m][2] = R[2][m]
        a.y = 0.0f;                           // A[m][3] = 0 (K padding)
    }

    // ---- Phase 1: WMMA coordinate transform + LDS scatter -------------------
    // Each wave covers 4096 voxels: 256 iterations x 16 voxels (one WMMA each).
    const int n = lane & 15;                  // this lane's voxel column
    for (int it = 0; it < NVOX / (8 * 16); ++it) {
        const int vbase = wave * (NVOX / 8) + it * 16;
        const int v  = vbase + n;
        const float sx = (float)((v >> 10) & 31) - 16.0f;
        const float sy = (float)((v >>  5) & 31) - 16.0f;
        const float sz = (float)( v        & 31) - 16.0f;

        // B layout (4x16): lanes 0-15 -> col n, VGPR0=K0, VGPR1=K1;
        //                  lanes 16-31 -> col n, VGPR0=K2, VGPR1=K3.
        v2f bb;
        bb.x = (lane < 16) ? sx : sz;
        bb.y = (lane < 16) ? sy : 0.0f;       // K=3 pad row

        v8f c = {0.f, 0.f, 0.f, 0.f, 0.f, 0.f, 0.f, 0.f};
        // D[m][n] = sum_k R[k][m] * scaled_k(voxel n)  (EXEC all-1 here)
        v8f d = __builtin_amdgcn_wmma_f32_16x16x4_f32(
            /*neg_a=*/false, a, /*neg_b=*/false, bb,
            /*c_mod=*/(short)0, c, /*reuse_a=*/false, /*reuse_b=*/false);

        if (lane < 16) {                      // lanes 16-31 hold D rows 8..15 (all zero)
            float cx = fminf(fmaxf(d[0] + 20.0f, 0.0f), (float)(GRID_N - 1));
            float cy = fminf(fmaxf(d[1] + 20.0f, 0.0f), (float)(GRID_N - 1));
            float cz = fminf(fmaxf(d[2] + 20.0f, 0.0f), (float)(GRID_N - 1));
            int ix = (int)cx, iy = (int)cy, iz = (int)cz;
            int g  = ix * (GRID_N * GRID_N) + iy * GRID_N + iz;
            st[g] = (unsigned short)(v + 1);  // racy last-write-wins == scatter .set
        }
    }
    __syncthreads();

    // ---- Phase 2: alpha-composite ray march along depth ---------------------
    for (int r = tid; r < GRID_N * GRID_N; r += BLOCK) {
        const int x = r / GRID_N;
        const int y = r % GRID_N;
        const int base = x * (GRID_N * GRID_N) + y * GRID_N;
        float T = 1.0f, acc = 0.0f;
        #pragma unroll 4
        for (int k = 0; k < GRID_N; ++k) {
            unsigned s = st[base + k];
            if (s) {                          // empty cell: sigmoid(-1e9)=0 -> no-op
                int vi = (int)s - 1;
                float tl = attn[vi];
                float al = absb[vi];
                float t  = 1.0f / (1.0f + expf(-tl));
                float av = 1.0f / (1.0f + expf(-al));
                acc += T * t * av;
                T   *= (1.0f - t);
            }
        }
        out[b * (GRID_N * GRID_N) + r] = acc;
    }
}

extern "C" void kernel_launch(void* const* d_in, const int* in_sizes, int n_in,
                              void* d_out, int out_size, void* d_ws, size_t ws_size,
                              hipStream_t stream) {
    const float* camR = (const float*)d_in[0];   // [512,3,3]
    const float* absb = (const float*)d_in[1];   // [32,32,32,1]
    const float* attn = (const float*)d_in[2];   // [32,32,32,1]
    float* out = (float*)d_out;                  // [512,40,40,1]

    const int B = in_sizes[0] / 9;               // 512
    const size_t lds_bytes = GRID_SZ * sizeof(unsigned short);  // 128000 B (< 320 KB WGP LDS)
    DifferentiableRenderer_10471130268177_kernel<<<B, BLOCK, lds_bytes, stream>>>(
        camR, absb, attn, out);
}